// Message_Attention_57363583205514
// MI455X (gfx1250) — compile-verified
//
#include <hip/hip_runtime.h>
#include <math.h>

// Problem constants (match reference)
#define NN   2048      // N nodes
#define DD   128       // feature dim
#define EE   16384     // edges
#define OUTW (2 * EE)  // output row width = 32768

typedef __attribute__((ext_vector_type(2))) float v2f;
typedef __attribute__((ext_vector_type(8))) float v8f;

// ---------------------------------------------------------------------------
// Kernel 1: Xw = X @ W1^T  via V_WMMA_F32_16X16X4_F32 (full f32 precision).
// One wave per 16x16 output tile; K-loop over 128 in steps of 4 (32 WMMAs).
// A frag (16x4, MxK): lane l -> M = l&15, K-pair = (l>>4)*2 .. +1  (float2)
// B frag (4x16, KxN): lane l -> N = l&15, K-pair = (l>>4)*2 .. +1  (float2)
//   B[k][n] = W1T[d+k][n] = W1[n_base+n][d+k]  -> same addressing as A on W1.
// C/D (16x16): VGPR v -> row (v + 8*(l>>4)), col (l&15).
// ---------------------------------------------------------------------------
__global__ __launch_bounds__(256) void gemm_xw_wmma(
    const float* __restrict__ X, const float* __restrict__ W1,
    float* __restrict__ Xw) {
  const int wave = (blockIdx.x * blockDim.x + threadIdx.x) >> 5;  // 0..1023
  const int lane = threadIdx.x & 31;
  const int n_tile = wave & 7;    // 8 tiles along the 128-wide output dim
  const int m_tile = wave >> 3;   // 128 tiles along the 2048 rows
  const int m_base = m_tile << 4;
  const int n_base = n_tile << 4;
  const int lhalf  = lane >> 4;   // 0 or 1 -> selects K pair {0,1} or {2,3}
  const int lmod   = lane & 15;

  const float* arow = X  + (size_t)(m_base + lmod) * DD + (lhalf << 1);
  const float* brow = W1 + (size_t)(n_base + lmod) * DD + (lhalf << 1);

  v8f c = {};
#pragma unroll 4
  for (int d = 0; d < DD; d += 4) {
    v2f a = *(const v2f*)(arow + d);   // 8B aligned: offsets are even floats
    v2f b = *(const v2f*)(brow + d);
    c = __builtin_amdgcn_wmma_f32_16x16x4_f32(
        /*neg_a=*/false, a, /*neg_b=*/false, b,
        /*c_mod=*/(short)0, c, /*reuse_a=*/false, /*reuse_b=*/false);
  }
#pragma unroll
  for (int v = 0; v < 8; ++v) {
    const int row = m_base + v + (lhalf << 3);
    Xw[(size_t)row * DD + n_base + lmod] = c[v];
  }
}

// ---------------------------------------------------------------------------
// Kernel 2: s_i[n] = Xw[n,:] . W2[0,0:128]; s_j[n] = Xw[n,:] . W2[0,128:256]
// One wave per row; shuffle tree reduction (wave32).
// ---------------------------------------------------------------------------
__global__ __launch_bounds__(256) void score_kernel(
    const float* __restrict__ Xw, const float* __restrict__ W2,
    float* __restrict__ s_i, float* __restrict__ s_j) {
  const int wave = (blockIdx.x * blockDim.x + threadIdx.x) >> 5;
  const int lane = threadIdx.x & 31;
  if (wave >= NN) return;
  const float* row = Xw + (size_t)wave * DD;
  float pa = 0.f, pb = 0.f;
#pragma unroll
  for (int k = lane; k < DD; k += 32) {
    const float x = row[k];
    pa = fmaf(x, W2[k], pa);
    pb = fmaf(x, W2[DD + k], pb);
  }
#pragma unroll
  for (int off = 16; off > 0; off >>= 1) {
    pa += __shfl_xor(pa, off, 32);
    pb += __shfl_xor(pb, off, 32);
  }
  if (lane == 0) { s_i[wave] = pa; s_j[wave] = pb; }
}

// ---------------------------------------------------------------------------
// Kernel 3: per-row masked-softmax statistics.
//   M[i] = max_{j: adj[i,j]>0} (s_i[i] + s_j[j]);  Z[i] = sum exp(v - M[i])
// Online max/sum per lane over 64 strided j's, then shuffle combine.
// Reads adj exactly once (16 MB total).
// ---------------------------------------------------------------------------
__global__ __launch_bounds__(256) void rowstats_kernel(
    const int* __restrict__ adj, const float* __restrict__ s_i,
    const float* __restrict__ s_j, float* __restrict__ M,
    float* __restrict__ Z) {
  const int wave = (blockIdx.x * blockDim.x + threadIdx.x) >> 5;
  const int lane = threadIdx.x & 31;
  if (wave >= NN) return;
  const float si = s_i[wave];
  const int* arow = adj + (size_t)wave * NN;

  float m = -INFINITY, z = 0.f;
  for (int j = lane; j < NN; j += 32) {
    if (arow[j] > 0) {
      const float v = si + s_j[j];
      if (v > m) { z = z * expf(m - v) + 1.f; m = v; }  // expf(-inf)=0 safe
      else       { z += expf(v - m); }
    }
  }
#pragma unroll
  for (int off = 16; off > 0; off >>= 1) {
    const float mo = __shfl_xor(m, off, 32);
    const float zo = __shfl_xor(z, off, 32);
    const float mn = fmaxf(m, mo);
    float zn = 0.f;
    if (z  > 0.f) zn += z  * expf(m  - mn);   // guards avoid (-inf)-(-inf)
    if (zo > 0.f) zn += zo * expf(mo - mn);
    m = mn; z = zn;
  }
  if (lane == 0) { M[wave] = m; Z[wave] = z; }
}

// ---------------------------------------------------------------------------
// Kernel 4: zero the 256 MB output with wide (B128) stores, grid-stride.
// This is the bandwidth-dominant step (~11 us at 23.3 TB/s).
// ---------------------------------------------------------------------------
__global__ __launch_bounds__(256) void zero_kernel(float4* __restrict__ out,
                                                   long n4) {
  long i = (long)blockIdx.x * blockDim.x + threadIdx.x;
  const long stride = (long)gridDim.x * blockDim.x;
  const float4 zq = make_float4(0.f, 0.f, 0.f, 0.f);
  for (; i < n4; i += stride) out[i] = zq;
}

// ---------------------------------------------------------------------------
// Kernel 5: scatter the 2E attention values.
//   out[dst, c]     = attention[dst, src]
//   out[src, c + E] = attention[src, dst]
// attention[i,j] = adj[i,j]>0 ? exp(s_i[i]+s_j[j] - M[i]) / Z[i] : 0
// ---------------------------------------------------------------------------
__global__ __launch_bounds__(256) void scatter_kernel(
    const long long* __restrict__ edges, const int* __restrict__ adj,
    const float* __restrict__ s_i, const float* __restrict__ s_j,
    const float* __restrict__ M, const float* __restrict__ Z,
    float* __restrict__ out) {
  const int c = blockIdx.x * blockDim.x + threadIdx.x;
  if (c >= EE) return;
  const int src = (int)edges[c];
  const int dst = (int)edges[EE + c];

  float ap = 0.f;
  if (adj[(size_t)dst * NN + src] > 0)
    ap = expf(s_i[dst] + s_j[src] - M[dst]) / Z[dst];
  out[(size_t)dst * OUTW + c] = ap;

  float ac = 0.f;
  if (adj[(size_t)src * NN + dst] > 0)
    ac = expf(s_i[src] + s_j[dst] - M[src]) / Z[src];
  out[(size_t)src * OUTW + EE + c] = ac;
}

// ---------------------------------------------------------------------------
extern "C" void kernel_launch(void* const* d_in, const int* in_sizes, int n_in,
                              void* d_out, int out_size, void* d_ws,
                              size_t ws_size, hipStream_t stream) {
  const float*      X     = (const float*)d_in[0];       // [N, D]
  const long long*  edges = (const long long*)d_in[1];   // [2, E] int64
  const int*        adj   = (const int*)d_in[2];         // [N, N]
  const float*      W1    = (const float*)d_in[3];       // [D, D]
  const float*      W2    = (const float*)d_in[4];       // [1, 2D]
  float* out = (float*)d_out;                            // [N, 2E]

  // Workspace layout (~1.03 MB)
  char* ws = (char*)d_ws;
  float* Xw  = (float*)ws;                                // 2048*128 f32
  float* s_i = (float*)(ws + (size_t)NN * DD * 4);        // 2048 f32
  float* s_j = s_i + NN;
  float* Mv  = s_j + NN;
  float* Zv  = Mv + NN;

  // 1) Xw = X @ W1^T : 1024 waves (128 m-tiles x 8 n-tiles), 8 waves/block
  gemm_xw_wmma<<<128, 256, 0, stream>>>(X, W1, Xw);

  // 2) scores: one wave per row -> 2048 waves -> 256 blocks
  score_kernel<<<256, 256, 0, stream>>>(Xw, W2, s_i, s_j);

  // 3) row softmax stats: one wave per row
  rowstats_kernel<<<256, 256, 0, stream>>>(adj, s_i, s_j, Mv, Zv);

  // 4) zero 256 MB output (16,777,216 float4)
  const long n4 = (long)NN * OUTW / 4;
  zero_kernel<<<16384, 256, 0, stream>>>((float4*)out, n4);

  // 5) scatter 2E attention values
  scatter_kernel<<<(EE + 255) / 256, 256, 0, stream>>>(edges, adj, s_i, s_j,
                                                       Mv, Zv, out);
}